// DecoderBiRNN_42013370089579
// MI455X (gfx1250) — compile-verified
//
#include <hip/hip_runtime.h>
#include <hip/hip_bf16.h>
#include <math.h>

// ---------------------------------------------------------------------------
// DecoderBiRNN for MI455X (gfx1250): fp32 WMMA (V_WMMA_F32_16X16X4_F32)
//   B=128, H=1024, H2=512, V=32000, T=32
// ---------------------------------------------------------------------------

#define Bq   128
#define Hq   1024
#define H2q  512
#define Vq   32000
#define Tq   32

typedef __attribute__((ext_vector_type(2))) float v2f;
typedef __attribute__((ext_vector_type(8))) float v8f;

__device__ __forceinline__ v8f wmma_f32(v2f a, v2f b, v8f c) {
    // D = A(16x4) * B(4x16) + C(16x16), full fp32
    return __builtin_amdgcn_wmma_f32_16x16x4_f32(
        /*neg_a=*/false, a, /*neg_b=*/false, b,
        /*c_mod=*/(short)0, c, /*reuse_a=*/false, /*reuse_b=*/false);
}

// Async global -> LDS copy of 16 bytes per lane (ASYNCcnt-tracked; no VGPR data)
__device__ __forceinline__ void async_copy_b128(unsigned lds_off, const void* gaddr) {
    asm volatile("global_load_async_to_lds_b128 %0, %1, off"
                 :: "v"(lds_off), "v"(gaddr) : "memory");
}
__device__ __forceinline__ void wait_asynccnt0() {
    asm volatile("s_wait_asynccnt 0x0" ::: "memory");
}

// ---------------------------------------------------------------------------
// Kernel 1: enc[B,H2] = relu(hidden)[B,H] @ W_sq[H2,H]^T + b_sq
// grid (H2/16, B/16), block 32 (one wave per 16x16 tile)
// ---------------------------------------------------------------------------
__global__ void k_squeeze(const float* __restrict__ hidden,
                          const float* __restrict__ W_sq,
                          const float* __restrict__ b_sq,
                          float* __restrict__ enc) {
    const int lane = threadIdx.x & 31;
    const int half = lane >> 4;      // 0: K=0,1  1: K=2,3
    const int l16  = lane & 15;
    const int n0 = blockIdx.x * 16;  // H2 tile
    const int m0 = blockIdx.y * 16;  // B tile

    v8f acc = {};
    const float* ap = hidden + (size_t)(m0 + l16) * Hq + 2 * half;
    const float* bp = W_sq   + (size_t)(n0 + l16) * Hq + 2 * half;
    #pragma unroll 8
    for (int k = 0; k < Hq; k += 4) {
        v2f a, b;
        a.x = fmaxf(ap[k],     0.0f);   // fused relu
        a.y = fmaxf(ap[k + 1], 0.0f);
        b.x = bp[k];
        b.y = bp[k + 1];
        acc = wmma_f32(a, b, acc);
    }
    const float bias = b_sq[n0 + l16];
    #pragma unroll
    for (int r = 0; r < 8; ++r) {
        const int m = m0 + r + 8 * half;
        enc[(size_t)m * H2q + n0 + l16] = acc[r] + bias;
    }
}

// ---------------------------------------------------------------------------
// Kernel 2: one GRU step, both directions (blockIdx.z = dir)
//   gh = h @ W_hh^T + b_hh ; r,z = sigmoid(b_ih + gh) ; n = tanh(b_ih + r*gh)
//   h' = (1-z)*n + z*h ; store h' and scatter into hcat[t or T-1-t]
// grid (H2/16, B/16, 2), block 32
// ---------------------------------------------------------------------------
__global__ void k_gru_step(const float* __restrict__ hF_in,
                           const float* __restrict__ hR_in,
                           const float* __restrict__ Wf,
                           const float* __restrict__ bihf,
                           const float* __restrict__ bhhf,
                           const float* __restrict__ Wr,
                           const float* __restrict__ bihr,
                           const float* __restrict__ bhhr,
                           float* __restrict__ hF_out,
                           float* __restrict__ hR_out,
                           float* __restrict__ hcat,
                           int t) {
    const int dir = blockIdx.z;
    const float* h_in = dir ? hR_in : hF_in;
    const float* W    = dir ? Wr    : Wf;
    const float* b_ih = dir ? bihr  : bihf;
    const float* b_hh = dir ? bhhr  : bhhf;
    float* h_out      = dir ? hR_out : hF_out;

    const int lane = threadIdx.x & 31;
    const int half = lane >> 4;
    const int l16  = lane & 15;
    const int n0 = blockIdx.x * 16;
    const int m0 = blockIdx.y * 16;

    v8f accR = {}, accZ = {}, accN = {};
    const float* ap  = h_in + (size_t)(m0 + l16) * H2q + 2 * half;
    const float* bpR = W    + (size_t)(n0 + l16) * H2q + 2 * half;
    const float* bpZ = bpR + (size_t)H2q * H2q;
    const float* bpN = bpZ + (size_t)H2q * H2q;
    #pragma unroll 4
    for (int k = 0; k < H2q; k += 4) {
        v2f a;  a.x  = ap[k];      a.y  = ap[k + 1];
        v2f br; br.x = bpR[k];     br.y = bpR[k + 1];
        v2f bz; bz.x = bpZ[k];     bz.y = bpZ[k + 1];
        v2f bn; bn.x = bpN[k];     bn.y = bpN[k + 1];
        accR = wmma_f32(a, br, accR);   // shared A fragment, 3 gates
        accZ = wmma_f32(a, bz, accZ);
        accN = wmma_f32(a, bn, accN);
    }

    const int nc   = n0 + l16;
    const int tS   = dir ? (Tq - 1 - t) : t;  // reverse scan stored flipped
    const int coff = dir ? H2q : 0;           // rev occupies cols [512,1024)
    const float bhh_r = b_hh[nc];
    const float bhh_z = b_hh[nc + H2q];
    const float bhh_n = b_hh[nc + 2 * H2q];
    const float bih_r = b_ih[nc];
    const float bih_z = b_ih[nc + H2q];
    const float bih_n = b_ih[nc + 2 * H2q];
    #pragma unroll
    for (int r = 0; r < 8; ++r) {
        const int m = m0 + r + 8 * half;
        const float gr = accR[r] + bhh_r;
        const float gz = accZ[r] + bhh_z;
        const float gn = accN[r] + bhh_n;
        const float rg = 1.0f / (1.0f + expf(-(bih_r + gr)));
        const float zg = 1.0f / (1.0f + expf(-(bih_z + gz)));
        const float ng = tanhf(bih_n + rg * gn);
        const float hp = h_in[(size_t)m * H2q + nc];
        const float hn = (1.0f - zg) * ng + zg * hp;
        h_out[(size_t)m * H2q + nc] = hn;
        hcat[(size_t)tS * (Bq * Hq) + (size_t)m * Hq + coff + nc] = hn;
    }
}

// ---------------------------------------------------------------------------
// Kernel 3: logits[4096, 32000] = hcat[4096,1024] @ W_out[32000,1024]^T + b_out
// grid (V/128, 4096/64), block 256 (8 waves)
// Block tile M=64 x N=128; each wave computes 4 vertical 16x16 subtiles so
// every B fragment (W_out) is reused 4x. A chunks (64x64) staged in LDS via
// GLOBAL_LOAD_ASYNC_TO_LDS_B128 (ASYNCcnt); next W_out chunk prefetched.
// ---------------------------------------------------------------------------
__global__ __launch_bounds__(256) void k_out_gemm(const float* __restrict__ hcat,
                                                  const float* __restrict__ W_out,
                                                  const float* __restrict__ b_out,
                                                  float* __restrict__ out) {
    __shared__ float As[64][68];  // 64 rows x 64-K chunk, padded (272B rows, 16B-aligned)

    const int wave = threadIdx.x >> 5;
    const int lane = threadIdx.x & 31;
    const int half = lane >> 4;
    const int l16  = lane & 15;
    const int m0 = blockIdx.y * 64;
    const int n0 = blockIdx.x * 128 + wave * 16;

    const int row16 = threadIdx.x >> 4;        // 0..15
    const int col4  = (threadIdx.x & 15) * 4;  // 0..60

    v8f acc[4] = {};
    const float* bp = W_out + (size_t)(n0 + l16) * Hq + 2 * half;

    for (int kc = 0; kc < Hq; kc += 64) {
        // async stage of A[64][64] (shared by all 8 waves); 16B per lane x4
        #pragma unroll
        for (int rr = 0; rr < 4; ++rr) {
            const int row = row16 + rr * 16;
            const unsigned lds_off =
                (unsigned)(unsigned long long)&As[row][col4];
            const float* src = hcat + (size_t)(m0 + row) * Hq + kc + col4;
            async_copy_b128(lds_off, src);
        }
        if (kc + 64 < Hq)
            __builtin_prefetch(bp + kc + 64, 0, 1);  // next W_out chunk -> global_prefetch_b8
        wait_asynccnt0();
        __syncthreads();

        #pragma unroll
        for (int kk = 0; kk < 64; kk += 4) {
            v2f b;
            const float* bpp = bp + kc + kk;
            b.x = bpp[0];
            b.y = bpp[1];
            #pragma unroll
            for (int s = 0; s < 4; ++s) {       // 4x M reuse of B fragment
                v2f a;
                a.x = As[l16 + 16 * s][kk + 2 * half];
                a.y = As[l16 + 16 * s][kk + 2 * half + 1];
                acc[s] = wmma_f32(a, b, acc[s]);
            }
        }
        __syncthreads();
    }

    const float bias = b_out[n0 + l16];
    #pragma unroll
    for (int s = 0; s < 4; ++s) {
        #pragma unroll
        for (int r = 0; r < 8; ++r) {
            const int m = m0 + 16 * s + r + 8 * half;
            out[(size_t)m * Vq + n0 + l16] = acc[s][r] + bias;
        }
    }
}

// ---------------------------------------------------------------------------
// Kernel 4: in-place log_softmax over V per row; row cached in 128KB LDS
// grid 4096, block 256, dynamic LDS = (V + 32) floats
// ---------------------------------------------------------------------------
__global__ __launch_bounds__(256) void k_logsoftmax(float* __restrict__ out) {
    extern __shared__ float smem[];
    float* rowbuf  = smem;
    float* scratch = smem + Vq;

    const int tid = threadIdx.x;
    float* p = out + (size_t)blockIdx.x * Vq;

    // pass 1: load row into LDS, local max
    float lmax = -INFINITY;
    for (int i = tid; i < Vq; i += 256) {
        const float v = p[i];
        rowbuf[i] = v;
        lmax = fmaxf(lmax, v);
    }
    #pragma unroll
    for (int off = 16; off > 0; off >>= 1)
        lmax = fmaxf(lmax, __shfl_xor(lmax, off, 32));
    if ((tid & 31) == 0) scratch[tid >> 5] = lmax;
    __syncthreads();
    float rmax = scratch[0];
    #pragma unroll
    for (int i = 1; i < 8; ++i) rmax = fmaxf(rmax, scratch[i]);
    __syncthreads();

    // pass 2: sum of exp(x - max) from LDS
    float lsum = 0.0f;
    for (int i = tid; i < Vq; i += 256)
        lsum += expf(rowbuf[i] - rmax);
    #pragma unroll
    for (int off = 16; off > 0; off >>= 1)
        lsum += __shfl_xor(lsum, off, 32);
    if ((tid & 31) == 0) scratch[tid >> 5] = lsum;
    __syncthreads();
    float rsum = 0.0f;
    #pragma unroll
    for (int i = 0; i < 8; ++i) rsum += scratch[i];
    const float lz = logf(rsum);

    // pass 3: write normalized
    for (int i = tid; i < Vq; i += 256)
        p[i] = rowbuf[i] - rmax - lz;
}

// ---------------------------------------------------------------------------
// Host launcher
// ---------------------------------------------------------------------------
extern "C" void kernel_launch(void* const* d_in, const int* in_sizes, int n_in,
                              void* d_out, int out_size, void* d_ws, size_t ws_size,
                              hipStream_t stream) {
    const float* hidden  = (const float*)d_in[0];
    const float* W_sq    = (const float*)d_in[1];
    const float* b_sq    = (const float*)d_in[2];
    const float* W_hh_f  = (const float*)d_in[3];
    const float* b_ih_f  = (const float*)d_in[4];
    const float* b_hh_f  = (const float*)d_in[5];
    const float* W_hh_r  = (const float*)d_in[6];
    const float* b_ih_r  = (const float*)d_in[7];
    const float* b_hh_r  = (const float*)d_in[8];
    const float* W_out   = (const float*)d_in[9];
    const float* b_out   = (const float*)d_in[10];
    float* out = (float*)d_out;

    // workspace layout (floats): enc | hA_f | hB_f | hA_r | hB_r | hcat
    float* ws   = (float*)d_ws;
    float* enc  = ws;
    float* hA_f = enc  + (size_t)Bq * H2q;
    float* hB_f = hA_f + (size_t)Bq * H2q;
    float* hA_r = hB_f + (size_t)Bq * H2q;
    float* hB_r = hA_r + (size_t)Bq * H2q;
    float* hcat = hB_r + (size_t)Bq * H2q;  // [T, B, H] = 4,194,304 floats

    // Phase 1: squeeze
    k_squeeze<<<dim3(H2q / 16, Bq / 16), 32, 0, stream>>>(hidden, W_sq, b_sq, enc);

    // Phase 2: 32 sequential GRU steps, both directions per launch
    const float* inF = enc;
    const float* inR = enc;
    float* outF = hA_f;
    float* outR = hA_r;
    for (int t = 0; t < Tq; ++t) {
        k_gru_step<<<dim3(H2q / 16, Bq / 16, 2), 32, 0, stream>>>(
            inF, inR, W_hh_f, b_ih_f, b_hh_f, W_hh_r, b_ih_r, b_hh_r,
            outF, outR, hcat, t);
        inF = outF;
        inR = outR;
        outF = (outF == hA_f) ? hB_f : hA_f;
        outR = (outR == hA_r) ? hB_r : hA_r;
    }

    // Phase 3: output projection (raw logits into d_out)
    k_out_gemm<<<dim3(Vq / 128, (Tq * Bq) / 64), 256, 0, stream>>>(
        hcat, W_out, b_out, out);

    // Phase 4: in-place log_softmax per row (128KB LDS row cache)
    k_logsoftmax<<<dim3(Tq * Bq), 256, (Vq + 32) * sizeof(float), stream>>>(out);
}